// FreeYOLOv2_10565619548932
// MI455X (gfx1250) — compile-verified
//
#include <hip/hip_runtime.h>
#include <hip/hip_bf16.h>
#include <math.h>

// ---------------- constants ----------------
#define NCLS   80
#define TOPKN  1000
#define NSEL   3000
#define NPAD   3008          // padded to 16 for WMMA row tiles
#define NWORDS 96            // 3072/32 (>= ceil(3008/32) = 94)
#define NBINS  4096
#define CANDCAP 4096
#define CONF_T 0.05f
#define NMS_T  0.6f
#define CLS_OFF 1e5f

typedef float v2f __attribute__((ext_vector_type(2)));
typedef float v8f __attribute__((ext_vector_type(8)));

// ---------------- workspace layout (bytes) ----------------
static constexpr size_t OFF_HIST = 0;                                  // 3*4096*4
static constexpr size_t OFF_CTR  = OFF_HIST + 3ull*NBINS*4;            // 3*4 (pad 64)
static constexpr size_t OFF_TH   = OFF_CTR + 64;                       // 3*4 (pad 64)
static constexpr size_t OFF_CVAL = OFF_TH + 64;                        // 3*4096*4
static constexpr size_t OFF_CIDX = OFF_CVAL + 3ull*CANDCAP*4;          // 3*4096*4
static constexpr size_t OFF_SEL_SCORE = OFF_CIDX + 3ull*CANDCAP*4;     // NPAD*4
static constexpr size_t OFF_SEL_LABEL = OFF_SEL_SCORE + NPAD*4;        // NPAD*4
static constexpr size_t OFF_SEL_VALID = OFF_SEL_LABEL + NPAD*4;        // NPAD*4
static constexpr size_t OFF_SEL_BOX   = OFF_SEL_VALID + NPAD*4;        // NPAD*16
static constexpr size_t OFF_SRT_SCORE = OFF_SEL_BOX + NPAD*16;
static constexpr size_t OFF_SRT_LABEL = OFF_SRT_SCORE + NPAD*4;
static constexpr size_t OFF_SRT_VALID = OFF_SRT_LABEL + NPAD*4;
static constexpr size_t OFF_SRT_BOX   = OFF_SRT_VALID + NPAD*4;
static constexpr size_t OFF_SMAT      = OFF_SRT_BOX + NPAD*16;         // NPAD*96*4
static constexpr size_t OFF_KEEP      = OFF_SMAT + (size_t)NPAD*NWORDS*4;

// ---------------- CDNA5 async global->LDS staging ----------------
__device__ __forceinline__ void async_copy_b128(unsigned lds_off, const float* gptr) {
  asm volatile("global_load_async_to_lds_b128 %0, %1, off"
               :: "v"(lds_off), "v"(gptr) : "memory");
}
__device__ __forceinline__ void wait_async0() {
  asm volatile("s_wait_asynccnt 0" ::: "memory");
}
__device__ __forceinline__ void wait_async1() {
  asm volatile("s_wait_asynccnt 1" ::: "memory");
}
__device__ __forceinline__ void wait_ds0() {
  asm volatile("s_wait_dscnt 0" ::: "memory");
}

// monotonic float -> uint key, top 12 bits as bin
__device__ __forceinline__ unsigned binOf(float x) {
  unsigned u = __float_as_uint(x);
  u = (u & 0x80000000u) ? ~u : (u | 0x80000000u);
  return u >> 20;
}

// ---------------- 1) zero hist/ctr ----------------
__global__ void k_init(unsigned* p, int n) {
  int i = blockIdx.x * blockDim.x + threadIdx.x;
  if (i < n) p[i] = 0u;
}

// ---------------- 2) streaming histogram, double-buffered async staging ----------------
#define HB 256
__global__ __launch_bounds__(HB) void k_hist(const float* __restrict__ cls,
                                             unsigned n, unsigned* __restrict__ ghist) {
  __shared__ float stage[2][HB * 4];
  __shared__ unsigned lhist[NBINS];
  const int tid = threadIdx.x;
  for (int b = tid; b < NBINS; b += HB) lhist[b] = 0u;
  __syncthreads();

  const unsigned chunk = HB * 4;
  const unsigned long long step = (unsigned long long)gridDim.x * chunk;
  unsigned long long base = (unsigned long long)blockIdx.x * chunk;

  auto stage_chunk = [&](unsigned long long b, int sb) {
    unsigned idx = (unsigned)b + (unsigned)tid * 4u;
    if (idx + 4u <= n) {
      unsigned ldsoff = (unsigned)(unsigned long long)(&stage[sb][tid * 4]);
      async_copy_b128(ldsoff, cls + idx);   // ASYNCcnt-tracked DMA into LDS
    } else {
      for (int k2 = 0; k2 < 4; ++k2) {
        unsigned j = idx + (unsigned)k2;
        stage[sb][tid * 4 + k2] = (j < n) ? cls[j] : -1e30f;
      }
    }
  };

  if (base < n) stage_chunk(base, 0);          // prologue
  int buf = 0;
  for (; base < n; base += step, buf ^= 1) {
    unsigned long long nxt = base + step;
    bool more = (nxt < n);
    if (more) {                                // prefetch next chunk into other buffer
      wait_ds0();                              // buffer recycle: own LDS reads/atomics done
      stage_chunk(nxt, buf ^ 1);
      wait_async1();                           // current chunk complete, next in flight
    } else {
      wait_async0();
    }
    unsigned idx = (unsigned)base + (unsigned)tid * 4u;
#pragma unroll
    for (int k2 = 0; k2 < 4; ++k2) {
      unsigned j = idx + (unsigned)k2;
      if (j < n) atomicAdd(&lhist[binOf(stage[buf][tid * 4 + k2])], 1u);
    }
  }
  __syncthreads();
  for (int b = tid; b < NBINS; b += HB)
    if (lhist[b]) atomicAdd(&ghist[b], lhist[b]);
}

// ---------------- 3) per-level threshold bin (cum from top >= TOPK) ----------------
__global__ __launch_bounds__(256) void k_thresh(const unsigned* __restrict__ hist,
                                                unsigned* __restrict__ th) {
  const int lvl = blockIdx.x;
  const unsigned* h = hist + (size_t)lvl * NBINS;
  __shared__ unsigned hh[NBINS];
  __shared__ unsigned part[256];
  const int tid = threadIdx.x;
  for (int b = tid; b < NBINS; b += 256) hh[b] = h[b];
  __syncthreads();
  unsigned s = 0;
  for (int i = 0; i < 16; ++i) s += hh[tid * 16 + i];
  part[tid] = s;
  __syncthreads();
  if (tid == 0) {
    unsigned cum = 0; int res = 0; bool found = false;
    for (int c = 255; c >= 0 && !found; --c) {
      if (cum + part[c] >= TOPKN) {
        unsigned cc = cum;
        for (int b = c * 16 + 15; b >= c * 16; --b) {
          cc += hh[b];
          if (cc >= TOPKN) { res = b; found = true; break; }
        }
        if (!found) { res = c * 16; found = true; }
      } else cum += part[c];
    }
    th[lvl] = (unsigned)res;
  }
}

// ---------------- 4) collect candidates >= threshold bin (async staged) ----------------
__global__ __launch_bounds__(HB) void k_collect(const float* __restrict__ cls, unsigned n,
                                                const unsigned* __restrict__ thp,
                                                unsigned* ctr, float* cval, unsigned* cidx) {
  __shared__ float stage[2][HB * 4];
  const int tid = threadIdx.x;
  const unsigned th = *thp;

  const unsigned chunk = HB * 4;
  const unsigned long long step = (unsigned long long)gridDim.x * chunk;
  unsigned long long base = (unsigned long long)blockIdx.x * chunk;

  auto stage_chunk = [&](unsigned long long b, int sb) {
    unsigned idx = (unsigned)b + (unsigned)tid * 4u;
    if (idx + 4u <= n) {
      unsigned ldsoff = (unsigned)(unsigned long long)(&stage[sb][tid * 4]);
      async_copy_b128(ldsoff, cls + idx);
    } else {
      for (int k2 = 0; k2 < 4; ++k2) {
        unsigned j = idx + (unsigned)k2;
        stage[sb][tid * 4 + k2] = (j < n) ? cls[j] : -1e30f;
      }
    }
  };

  if (base < n) stage_chunk(base, 0);
  int buf = 0;
  for (; base < n; base += step, buf ^= 1) {
    unsigned long long nxt = base + step;
    bool more = (nxt < n);
    if (more) {
      wait_ds0();
      stage_chunk(nxt, buf ^ 1);
      wait_async1();
    } else {
      wait_async0();
    }
    unsigned idx = (unsigned)base + (unsigned)tid * 4u;
#pragma unroll
    for (int k2 = 0; k2 < 4; ++k2) {
      unsigned j = idx + (unsigned)k2;
      if (j < n) {
        float x = stage[buf][tid * 4 + k2];
        if (binOf(x) >= th) {
          unsigned slot = atomicAdd(ctr, 1u);
          if (slot < CANDCAP) { cval[slot] = x; cidx[slot] = j; }
        }
      }
    }
  }
}

// ---------------- bitonic sort of 4096 (desc by val, asc by idx) ----------------
__device__ void bitonic4096_desc(float* sv, unsigned* si, int tid) {
  for (unsigned k = 2; k <= 4096u; k <<= 1) {
    for (unsigned j = k >> 1; j > 0; j >>= 1) {
      __syncthreads();
      for (unsigned t = (unsigned)tid; t < 4096u; t += 1024u) {
        unsigned x = t ^ j;
        if (x > t) {
          float va = sv[t], vb = sv[x];
          unsigned ia = si[t], ib = si[x];
          bool a_before = (va > vb) || (va == vb && ia < ib);  // strict desc order
          bool dirDesc = ((t & k) == 0u);
          if (dirDesc ? !a_before : a_before) {
            sv[t] = vb; sv[x] = va; si[t] = ib; si[x] = ia;
          }
        }
      }
    }
  }
  __syncthreads();
}

// ---------------- 5) per-level sort + top-1000 decode ----------------
__global__ __launch_bounds__(1024) void k_sort_level(
    const float* __restrict__ cval, const unsigned* __restrict__ cidx,
    const unsigned* __restrict__ ctr, const float* __restrict__ reg,
    int W, float strd, int M,
    float* sel_score, int* sel_label, unsigned* sel_valid, float* sel_box, int lvlOff) {
  __shared__ float sv[4096];
  __shared__ unsigned si[4096];
  const int tid = threadIdx.x;
  unsigned cnt = *ctr; if (cnt > (unsigned)CANDCAP) cnt = CANDCAP;
  for (int t = tid; t < 4096; t += 1024) {
    if ((unsigned)t < cnt) { sv[t] = cval[t]; si[t] = cidx[t]; }
    else                   { sv[t] = -3.4e38f; si[t] = 0xFFFFFFFFu; }
  }
  bitonic4096_desc(sv, si, tid);
  if (tid < TOPKN) {
    float lv = sv[tid];
    unsigned fi = si[tid];
    if (fi >= (unsigned)M * (unsigned)NCLS) fi = 0u;  // pad safety
    int a = (int)(fi / NCLS);
    int lab = (int)(fi % NCLS);
    float ax = ((float)(a % W) + 0.5f) * strd;
    float ay = ((float)(a / W) + 0.5f) * strd;
    const float* r4 = reg + (size_t)a * 4;
    float cx = ax + r4[0] * strd, cy = ay + r4[1] * strd;
    float bw = expf(r4[2]) * strd, bh = expf(r4[3]) * strd;
    float score = 1.0f / (1.0f + expf(-lv));
    int g = lvlOff + tid;
    sel_score[g] = score;
    sel_label[g] = lab;
    sel_valid[g] = (score > CONF_T) ? 1u : 0u;
    sel_box[4 * g + 0] = cx - 0.5f * bw;
    sel_box[4 * g + 1] = cy - 0.5f * bh;
    sel_box[4 * g + 2] = cx + 0.5f * bw;
    sel_box[4 * g + 3] = cy + 0.5f * bh;
  }
}

// ---------------- 6) global sort of 3000 (stable by position) ----------------
__global__ __launch_bounds__(1024) void k_sort_global(
    const float* sel_score, const int* sel_label, const unsigned* sel_valid,
    const float* sel_box,
    float* srt_score, int* srt_label, unsigned* srt_valid, float* srt_box) {
  __shared__ float sv[4096];
  __shared__ unsigned si[4096];
  const int tid = threadIdx.x;
  for (int t = tid; t < 4096; t += 1024) {
    if (t < NSEL) { sv[t] = sel_score[t]; si[t] = (unsigned)t; }
    else          { sv[t] = -3.4e38f; si[t] = 0xFFFFFFFFu; }
  }
  bitonic4096_desc(sv, si, tid);
  for (int t = tid; t < NPAD; t += 1024) {
    if (t < NSEL) {
      unsigned p = si[t];
      srt_score[t] = sv[t];
      srt_label[t] = sel_label[p];
      srt_valid[t] = sel_valid[p];
      for (int c = 0; c < 4; ++c) srt_box[4 * t + c] = sel_box[4 * p + c];
    } else {
      srt_score[t] = 0.f; srt_label[t] = 0; srt_valid[t] = 0u;
      for (int c = 0; c < 4; ++c) srt_box[4 * t + c] = 0.f;
    }
  }
}

// ---------------- 7) pairwise IoU tiles: WMMA computes area[i]+area[j] ----------------
__global__ __launch_bounds__(32) void k_iou(const float* __restrict__ srt_box,
                                            const int* __restrict__ srt_label,
                                            unsigned* __restrict__ S) {
  __shared__ float rb4[16][4];
  __shared__ float cb4[32][4];
  __shared__ float rarea[16];
  __shared__ float carea[32];
  const int lane = threadIdx.x;
  const int i0 = blockIdx.x * 16, j0 = blockIdx.y * 32;
  if (lane < 16) {
    int i = i0 + lane;
    float off = (float)srt_label[i] * CLS_OFF;
    float x1 = srt_box[4 * i + 0] + off, y1 = srt_box[4 * i + 1] + off;
    float x2 = srt_box[4 * i + 2] + off, y2 = srt_box[4 * i + 3] + off;
    rb4[lane][0] = x1; rb4[lane][1] = y1; rb4[lane][2] = x2; rb4[lane][3] = y2;
    rarea[lane] = (x2 - x1) * (y2 - y1);
  }
  {
    int j = j0 + lane;
    float off = (float)srt_label[j] * CLS_OFF;
    float x1 = srt_box[4 * j + 0] + off, y1 = srt_box[4 * j + 1] + off;
    float x2 = srt_box[4 * j + 2] + off, y2 = srt_box[4 * j + 3] + off;
    cb4[lane][0] = x1; cb4[lane][1] = y1; cb4[lane][2] = x2; cb4[lane][3] = y2;
    carea[lane] = (x2 - x1) * (y2 - y1);
  }
  __syncthreads();

  // A (16x4): row m = [area_m, 1, 0, 0];  B (4x16): [ones; area_n; 0; 0]
  v2f A; A.x = (lane < 16) ? rarea[lane] : 0.f; A.y = (lane < 16) ? 1.f : 0.f;
  v2f B0; B0.x = (lane < 16) ? 1.f : 0.f; B0.y = (lane < 16) ? carea[lane] : 0.f;
  v2f B1; B1.x = (lane < 16) ? 1.f : 0.f; B1.y = (lane < 16) ? carea[lane + 16] : 0.f;
  v8f C = {0.f, 0.f, 0.f, 0.f, 0.f, 0.f, 0.f, 0.f};
  v8f d0 = __builtin_amdgcn_wmma_f32_16x16x4_f32(false, A, false, B0, (short)0, C, false, false);
  v8f d1 = __builtin_amdgcn_wmma_f32_16x16x4_f32(false, A, false, B1, (short)0, C, false, false);

  const int N = lane & 15;
  const int Mb = (lane < 16) ? 0 : 8;
  unsigned myword = 0u;
#pragma unroll
  for (int v = 0; v < 8; ++v) {
    int r = Mb + v;
    bool p0, p1;
    {
      int c = N;
      float xx1 = fmaxf(rb4[r][0], cb4[c][0]);
      float yy1 = fmaxf(rb4[r][1], cb4[c][1]);
      float xx2 = fminf(rb4[r][2], cb4[c][2]);
      float yy2 = fminf(rb4[r][3], cb4[c][3]);
      float inter = fmaxf(xx2 - xx1, 0.f) * fmaxf(yy2 - yy1, 0.f);
      // iou > T  <=>  inter > T*(areaSum - inter + eps)   (denom > 0 always)
      p0 = inter > NMS_T * (d0[v] - inter + 1e-9f);
    }
    {
      int c = 16 + N;
      float xx1 = fmaxf(rb4[r][0], cb4[c][0]);
      float yy1 = fmaxf(rb4[r][1], cb4[c][1]);
      float xx2 = fminf(rb4[r][2], cb4[c][2]);
      float yy2 = fminf(rb4[r][3], cb4[c][3]);
      float inter = fmaxf(xx2 - xx1, 0.f) * fmaxf(yy2 - yy1, 0.f);
      p1 = inter > NMS_T * (d1[v] - inter + 1e-9f);
    }
    unsigned long long bl0 = __ballot(p0);
    unsigned long long bl1 = __ballot(p1);
    unsigned lo = ((unsigned)bl0 & 0xFFFFu) | (((unsigned)bl1 & 0xFFFFu) << 16);
    unsigned hi = (((unsigned)(bl0 >> 16)) & 0xFFFFu) |
                  ((((unsigned)(bl1 >> 16)) & 0xFFFFu) << 16);
    if (lane == v)     myword = lo;  // row i0+v
    if (lane == v + 8) myword = hi;  // row i0+v+8
  }
  if (lane < 16) S[(size_t)(i0 + lane) * NWORDS + blockIdx.y] = myword;
}

// ---------------- 8) greedy NMS scan (one wave) ----------------
__global__ __launch_bounds__(32) void k_nms(const unsigned* __restrict__ S,
                                            const unsigned* __restrict__ valid,
                                            unsigned* __restrict__ keepArr) {
  __shared__ unsigned kw[NWORDS];
  const int lane = threadIdx.x;
  for (int w = lane; w < NWORDS; w += 32) kw[w] = 0u;
  __syncthreads();
  for (int i = 0; i < NSEL; ++i) {
    const int wfull = i >> 5;
    const unsigned rbits = (unsigned)(i & 31);
    unsigned acc = 0u;
    for (int wb = 0; wb < NWORDS; wb += 32) {
      int w = wb + lane;
      unsigned m = (w < wfull) ? 0xFFFFFFFFu
                 : ((w == wfull) ? (rbits ? ((1u << rbits) - 1u) : 0u) : 0u);
      if (m) acc |= (S[(size_t)i * NWORDS + w] & kw[w] & m);
    }
    bool sup = __any(acc != 0u);
    if (lane == 0) {
      unsigned k = (valid[i] && !sup) ? 1u : 0u;
      keepArr[i] = k;
      if (k) kw[wfull] |= (1u << rbits);
    }
    __syncthreads();
  }
}

// ---------------- 9) write outputs (boxes | scores | labels | keep) ----------------
__global__ void k_out(const float* __restrict__ srt_score, const int* __restrict__ srt_label,
                      const float* __restrict__ srt_box, const unsigned* __restrict__ keep,
                      float* __restrict__ out) {
  int i = blockIdx.x * blockDim.x + threadIdx.x;
  if (i >= NSEL) return;
  unsigned k = keep[i];
  for (int c = 0; c < 4; ++c)
    out[i * 4 + c] = k ? srt_box[i * 4 + c] : 0.f;
  out[12000 + i] = k ? srt_score[i] : 0.f;
  out[15000 + i] = k ? (float)srt_label[i] : -1.f;
  out[18000 + i] = k ? 1.f : 0.f;
}

// ---------------- launch ----------------
extern "C" void kernel_launch(void* const* d_in, const int* in_sizes, int n_in,
                              void* d_out, int out_size, void* d_ws, size_t ws_size,
                              hipStream_t stream) {
  (void)in_sizes; (void)n_in; (void)out_size; (void)ws_size;
  const float* cls[3] = {(const float*)d_in[0], (const float*)d_in[2], (const float*)d_in[4]};
  const float* reg[3] = {(const float*)d_in[1], (const float*)d_in[3], (const float*)d_in[5]};
  const int   Wl[3] = {512, 256, 128};
  const int   Ml[3] = {512 * 512, 256 * 256, 128 * 128};
  const float St[3] = {8.f, 16.f, 32.f};
  const unsigned Nl[3] = {(unsigned)Ml[0] * NCLS, (unsigned)Ml[1] * NCLS, (unsigned)Ml[2] * NCLS};
  const int histBlocks[3] = {2048, 1024, 512};

  char* ws = (char*)d_ws;
  unsigned* hist = (unsigned*)(ws + OFF_HIST);
  unsigned* ctr  = (unsigned*)(ws + OFF_CTR);
  unsigned* th   = (unsigned*)(ws + OFF_TH);
  float*    cval = (float*)(ws + OFF_CVAL);
  unsigned* cidx = (unsigned*)(ws + OFF_CIDX);
  float*    sel_score = (float*)(ws + OFF_SEL_SCORE);
  int*      sel_label = (int*)(ws + OFF_SEL_LABEL);
  unsigned* sel_valid = (unsigned*)(ws + OFF_SEL_VALID);
  float*    sel_box   = (float*)(ws + OFF_SEL_BOX);
  float*    srt_score = (float*)(ws + OFF_SRT_SCORE);
  int*      srt_label = (int*)(ws + OFF_SRT_LABEL);
  unsigned* srt_valid = (unsigned*)(ws + OFF_SRT_VALID);
  float*    srt_box   = (float*)(ws + OFF_SRT_BOX);
  unsigned* Smat      = (unsigned*)(ws + OFF_SMAT);
  unsigned* keepArr   = (unsigned*)(ws + OFF_KEEP);

  // zero hist + counters (contiguous region up to OFF_CVAL)
  const int zeroN = (int)((OFF_CVAL - OFF_HIST) / 4);
  k_init<<<(zeroN + 255) / 256, 256, 0, stream>>>((unsigned*)(ws + OFF_HIST), zeroN);

  for (int l = 0; l < 3; ++l)
    k_hist<<<histBlocks[l], HB, 0, stream>>>(cls[l], Nl[l], hist + (size_t)l * NBINS);

  k_thresh<<<3, 256, 0, stream>>>(hist, th);

  for (int l = 0; l < 3; ++l)
    k_collect<<<histBlocks[l], HB, 0, stream>>>(cls[l], Nl[l], th + l, ctr + l,
                                                cval + (size_t)l * CANDCAP,
                                                cidx + (size_t)l * CANDCAP);

  for (int l = 0; l < 3; ++l)
    k_sort_level<<<1, 1024, 0, stream>>>(cval + (size_t)l * CANDCAP,
                                         cidx + (size_t)l * CANDCAP, ctr + l,
                                         reg[l], Wl[l], St[l], Ml[l],
                                         sel_score, sel_label, sel_valid, sel_box,
                                         l * TOPKN);

  k_sort_global<<<1, 1024, 0, stream>>>(sel_score, sel_label, sel_valid, sel_box,
                                        srt_score, srt_label, srt_valid, srt_box);

  dim3 gi(NPAD / 16, NWORDS);  // 188 x 96 wave-sized tiles
  k_iou<<<gi, 32, 0, stream>>>(srt_box, srt_label, Smat);

  k_nms<<<1, 32, 0, stream>>>(Smat, srt_valid, keepArr);

  k_out<<<(NSEL + 255) / 256, 256, 0, stream>>>(srt_score, srt_label, srt_box, keepArr,
                                                (float*)d_out);
}